// BaseModel_74302934220896
// MI455X (gfx1250) — compile-verified
//
#include <hip/hip_runtime.h>
#include <hip/hip_bf16.h>
#include <math.h>

// ---------------------------------------------------------------------------
// Problem constants (match reference)
// ---------------------------------------------------------------------------
#define BB   2048      // batch
#define NN   20000     // n_tail
#define EE   512       // embed dim
#define CC   128       // class num
#define TOPK 10

typedef __attribute__((ext_vector_type(2))) float v2f;
typedef __attribute__((ext_vector_type(8))) float v8f;

// ---------------------------------------------------------------------------
// Tiled fp32 WMMA GEMM:  C[M,Nc] = act( A[M,K] @ B[K,Nc] + bias )
//   AMODE 0: A = A0 (row-major, lda)
//   AMODE 1: A = concat(A0[M,512], A1[M,512]) along K   (K == 1024)
//   AMODE 2: A[m, k] = A0[m, k>>7] * A1[m, k&127]       (outer product, C=128)
//   BT    0: B element = B[k*ldb + n]   (row-major [K,Nc])
//   BT    1: B element = B[n*ldb + k]   (row-major [Nc,K], i.e. @ B^T)
//
// Block: 128 threads (4 waves). Block tile 64x64, K-step 16.
// Each wave computes a 16x64 strip = 4 x v_wmma_f32_16x16x4_f32 accumulators.
// - LDS double-buffered; tile t+1 prefetched (global_load_b128) into registers
//   before the WMMAs of tile t: one barrier per K-tile, load latency hidden.
// - A/B stored K-pair-packed (v2f) in LDS: every fragment is one aligned
//   ds_load_b64 into an even VGPR pair (no repacking movs).
// - All 20 fragments of a tile are hoisted into registers before the 16-WMMA
//   burst so the XDL pipe runs back-to-back without dscnt stalls.
// Requirements relied on: K % 16 == 0, Nc % 4 == 0, lda/ldb % 4 == 0,
// 16B-aligned matrix bases (hipMalloc / 4-aligned float offsets).
// ---------------------------------------------------------------------------
#define BM  64
#define BN  64
#define BK  16
#define BKH (BK / 2)

template <int AMODE, int BT, bool RELU, bool BIAS>
__global__ __launch_bounds__(128)
void gemm_wmma_f32(const float* __restrict__ A0, const float* __restrict__ A1,
                   const float* __restrict__ Bm, const float* __restrict__ bias,
                   float* __restrict__ Cm,
                   int M, int Nc, int K, int lda, int ldb, int ldc)
{
    // K-pair packed tiles: element [r][p] = { X[r][2p], X[r][2p+1] }
    __shared__ v2f sA[2][BM][BKH + 1];   // +1 v2f pad: de-conflict row stride
    __shared__ v2f sB[2][BKH][BN + 1];

    const int tid  = threadIdx.x;       // 0..127
    const int wave = tid >> 5;          // 0..3
    const int lane = tid & 31;
    const int lo   = lane & 15;
    const int hi   = lane >> 4;
    const int brow = blockIdx.y * BM;
    const int bcol = blockIdx.x * BN;

    v8f acc[4] = {};                    // wave-private 16x64 strip

    // register staging for the next K-tile
    float4 a4[2];                       // AMODE 0/1: two 16B rowslices of A
    float  ar[8];                       // AMODE 2  : computed elements
    float4 b4[2];                       // BT0: rows {2p,2p+1} x 4 cols ; BT1: two 16B k-slices

    // ---- global -> regs for one K-tile (vectorized b128 where contiguous) ----
    auto load_tile = [&](int k0) {
        if (AMODE != 2) {
            #pragma unroll
            for (int j = 0; j < 2; ++j) {            // A tile: 64x16 = 256 float4
                const int idx = tid + j * 128;
                const int r   = idx >> 2;
                const int gk  = k0 + ((idx & 3) << 2);
                const int gr  = brow + r;
                float4 v = {0.0f, 0.0f, 0.0f, 0.0f};
                if (gr < M) {
                    const float* src =
                        (AMODE == 0) ? &A0[(size_t)gr * lda + gk]
                                     : ((gk < 512) ? &A0[(size_t)gr * 512 + gk]
                                                   : &A1[(size_t)gr * 512 + (gk - 512)]);
                    v = *(const float4*)src;
                }
                a4[j] = v;
            }
        } else {
            #pragma unroll
            for (int j = 0; j < 8; ++j) {            // outer product rs[b,k]*ha[b,c]
                const int i  = tid + j * 128;
                const int r  = i >> 4;
                const int kk = i & 15;
                const int gr = brow + r;
                const int gk = k0 + kk;
                ar[j] = (gr < M) ? A0[(size_t)gr * 128 + (gk >> 7)] *
                                   A1[(size_t)gr * 128 + (gk & 127)]
                                 : 0.0f;
            }
        }
        if (BT == 0) {                               // thread owns k-pair p x 4 cols
            const int p  = tid >> 4;                 // 0..7
            const int gc = bcol + ((tid & 15) << 2);
            float4 v0 = {0.0f, 0.0f, 0.0f, 0.0f}, v1 = v0;
            if (gc < Nc) {
                v0 = *(const float4*)&Bm[(size_t)(k0 + 2 * p)     * ldb + gc];
                v1 = *(const float4*)&Bm[(size_t)(k0 + 2 * p + 1) * ldb + gc];
            }
            b4[0] = v0; b4[1] = v1;
        } else {                                     // 16x64 = 256 float4 along k
            #pragma unroll
            for (int j = 0; j < 2; ++j) {
                const int idx = tid + j * 128;
                const int c   = idx >> 2;
                const int gk  = k0 + ((idx & 3) << 2);
                const int gc  = bcol + c;
                float4 v = {0.0f, 0.0f, 0.0f, 0.0f};
                if (gc < Nc) v = *(const float4*)&Bm[(size_t)gc * ldb + gk];
                b4[j] = v;
            }
        }
    };

    // ---- regs -> LDS (K-pair packed) ----
    auto store_tile = [&](int buf) {
        if (AMODE != 2) {
            #pragma unroll
            for (int j = 0; j < 2; ++j) {
                const int idx = tid + j * 128;
                const int r   = idx >> 2;
                const int kq  = idx & 3;
                v2f* dst = &sA[buf][r][kq << 1];
                v2f t0; t0.x = a4[j].x; t0.y = a4[j].y;
                v2f t1; t1.x = a4[j].z; t1.y = a4[j].w;
                dst[0] = t0; dst[1] = t1;
            }
        } else {
            #pragma unroll
            for (int j = 0; j < 8; ++j) {
                const int i  = tid + j * 128;
                const int r  = i >> 4;
                const int kk = i & 15;
                ((float*)&sA[buf][r][kk >> 1])[kk & 1] = ar[j];
            }
        }
        if (BT == 0) {
            const int p  = tid >> 4;
            const int c4 = (tid & 15) << 2;
            v2f t;
            t.x = b4[0].x; t.y = b4[1].x; sB[buf][p][c4 + 0] = t;
            t.x = b4[0].y; t.y = b4[1].y; sB[buf][p][c4 + 1] = t;
            t.x = b4[0].z; t.y = b4[1].z; sB[buf][p][c4 + 2] = t;
            t.x = b4[0].w; t.y = b4[1].w; sB[buf][p][c4 + 3] = t;
        } else {
            #pragma unroll
            for (int j = 0; j < 2; ++j) {
                const int idx = tid + j * 128;
                const int c   = idx >> 2;
                const int kq  = idx & 3;
                v2f t0; t0.x = b4[j].x; t0.y = b4[j].y;
                v2f t1; t1.x = b4[j].z; t1.y = b4[j].w;
                sB[buf][(kq << 1) + 0][c] = t0;
                sB[buf][(kq << 1) + 1][c] = t1;
            }
        }
    };

    // ---- software pipeline ----
    load_tile(0);
    store_tile(0);
    __syncthreads();

    const int nk   = K / BK;            // all K here are multiples of 16
    const int arow = (wave << 4) + lo;

    for (int t = 0; t < nk; ++t) {
        const int  cur  = t & 1;
        const bool more = (t + 1 < nk);
        if (more) load_tile((t + 1) * BK);   // global latency overlaps WMMAs below

        // hoist all fragments, then burst the XDL pipe
        v2f af[4], bf[16];
        #pragma unroll
        for (int q = 0; q < 4; ++q) {
            const int p = (q << 1) + hi;     // K-pair index for this half-wave
            af[q] = sA[cur][arow][p];
            #pragma unroll
            for (int u = 0; u < 4; ++u)
                bf[(q << 2) + u] = sB[cur][p][(u << 4) + lo];
        }
        #pragma unroll
        for (int q = 0; q < 4; ++q) {
            #pragma unroll
            for (int u = 0; u < 4; ++u) {
                acc[u] = __builtin_amdgcn_wmma_f32_16x16x4_f32(
                    /*neg_a=*/false, af[q], /*neg_b=*/false, bf[(q << 2) + u],
                    /*c_mod=*/(short)0, acc[u],
                    /*reuse_a=*/false, /*reuse_b=*/false);
            }
        }

        if (more) {
            store_tile(cur ^ 1);
            __syncthreads();
        }
    }

    // ---- epilogue: C/D layout = VGPR i -> row i (+8 for hi lanes), col = lane&15 ----
    const int rbase = brow + (wave << 4) + (hi << 3);
    #pragma unroll
    for (int u = 0; u < 4; ++u) {
        const int gc = bcol + (u << 4) + lo;
        if (gc >= Nc) continue;
        const float bv = BIAS ? bias[gc] : 0.0f;
        #pragma unroll
        for (int i = 0; i < 8; ++i) {
            const int gr = rbase + i;
            if (gr < M) {
                float v = acc[u][i] + bv;
                if (RELU) v = fmaxf(v, 0.0f);
                Cm[(size_t)gr * ldc + gc] = v;
            }
        }
    }
}

// ---------------------------------------------------------------------------
// Training-mode BatchNorm1d over columns (batch statistics, eps=1e-5).
// One block per column; in-place safe (all reads precede writes per column).
// ---------------------------------------------------------------------------
__global__ __launch_bounds__(256)
void bn_kernel(const float* __restrict__ x, float* __restrict__ y,
               const float* __restrict__ g, const float* __restrict__ b,
               int rows, int cols)
{
    const int c   = blockIdx.x;
    const int tid = threadIdx.x;
    __shared__ float s1[256];
    __shared__ float s2[256];

    float sum = 0.0f, sq = 0.0f;
    for (int r = tid; r < rows; r += 256) {
        const float v = x[(size_t)r * cols + c];
        sum += v; sq += v * v;
    }
    s1[tid] = sum; s2[tid] = sq;
    __syncthreads();
    for (int s = 128; s > 0; s >>= 1) {
        if (tid < s) { s1[tid] += s1[tid + s]; s2[tid] += s2[tid + s]; }
        __syncthreads();
    }
    const float mean = s1[0] / (float)rows;
    const float var  = s2[0] / (float)rows - mean * mean;
    const float inv  = rsqrtf(var + 1e-5f);
    const float gg = g[c], bb = b[c];
    for (int r = tid; r < rows; r += 256) {
        const size_t idx = (size_t)r * cols + c;
        y[idx] = gg * (x[idx] - mean) * inv + bb;
    }
}

// ---------------------------------------------------------------------------
// Soft top-k mask (k=10, temp=0.5) over rows of [rows, 128].
// thr = 10th largest (duplicates counted, matching lax.top_k) via 10 argmax
// removals; out = sigmoid((x - thr)/0.5) * x. In-place safe.
// ---------------------------------------------------------------------------
__global__ __launch_bounds__(128)
void topk_mask_kernel(const float* __restrict__ x, float* __restrict__ y)
{
    const int row = blockIdx.x;
    const int tid = threadIdx.x;          // 0..127 == column
    __shared__ float sv[128];
    __shared__ int   si[128];

    const size_t base = (size_t)row * CC;
    const float myv = x[base + tid];
    float work = myv;
    float thr  = 0.0f;

    for (int it = 0; it < TOPK; ++it) {
        sv[tid] = work; si[tid] = tid;
        __syncthreads();
        for (int s = 64; s > 0; s >>= 1) {
            if (tid < s) {
                const bool take = (sv[tid + s] > sv[tid]) ||
                                  (sv[tid + s] == sv[tid] && si[tid + s] < si[tid]);
                if (take) { sv[tid] = sv[tid + s]; si[tid] = si[tid + s]; }
            }
            __syncthreads();
        }
        thr = sv[0];
        if (tid == si[0]) work = -3.402823466e38f;   // remove exactly one instance
        __syncthreads();
    }

    const float z   = (myv - thr) * 2.0f;            // / TEMP(=0.5)
    const float sig = 1.0f / (1.0f + expf(-z));
    y[base + tid] = sig * myv;
}

// ---------------------------------------------------------------------------
// inter[b,d] = sum_c hrm[b,c] * tanh(codebook[rel[b], c, d])
// Fuses the gather + tanh + batched matvec (never materializes pc: saves 134MB).
// ---------------------------------------------------------------------------
__global__ __launch_bounds__(128)
void inter_kernel(const float* __restrict__ hrm, const int* __restrict__ rel,
                  const float* __restrict__ cb, float* __restrict__ inter)
{
    const int b = blockIdx.x;
    const int d = threadIdx.x;            // 0..127
    __shared__ float sh[128];
    sh[d] = hrm[(size_t)b * CC + d];
    __syncthreads();

    const float* m = cb + (size_t)rel[b] * CC * CC;
    float acc = 0.0f;
    #pragma unroll 4
    for (int c = 0; c < CC; ++c)
        acc += sh[c] * tanhf(m[(size_t)c * CC + d]);
    inter[(size_t)b * CC + d] = acc;
}

// ---------------------------------------------------------------------------
// Host-side launch
// ---------------------------------------------------------------------------
extern "C" void kernel_launch(void* const* d_in, const int* in_sizes, int n_in,
                              void* d_out, int out_size, void* d_ws, size_t ws_size,
                              hipStream_t stream)
{
    (void)in_sizes; (void)n_in; (void)out_size; (void)ws_size;

    const float* head = (const float*)d_in[0];
    const float* relv = (const float*)d_in[1];
    const int*   ridx = (const int*)d_in[2];
    const float* tail = (const float*)d_in[3];
    // d_in[4] = label (unused by forward)
    const float* codebook = (const float*)d_in[5];
    const float* core     = (const float*)d_in[6];
    const float* hsw1 = (const float*)d_in[7],  *hsb1 = (const float*)d_in[8];
    const float* hsw2 = (const float*)d_in[9],  *hsb2 = (const float*)d_in[10];
    const float* rsw1 = (const float*)d_in[11], *rsb1 = (const float*)d_in[12];
    const float* rsw2 = (const float*)d_in[13], *rsb2 = (const float*)d_in[14];
    const float* tsw1 = (const float*)d_in[15], *tsb1 = (const float*)d_in[16];
    const float* tsw2 = (const float*)d_in[17], *tsb2 = (const float*)d_in[18];
    const float* bn0g = (const float*)d_in[19], *bn0b = (const float*)d_in[20];
    const float* bn1g = (const float*)d_in[21], *bn1b = (const float*)d_in[22];
    const float* hrw1 = (const float*)d_in[23], *hrb1 = (const float*)d_in[24];
    const float* hrw2 = (const float*)d_in[25], *hrb2 = (const float*)d_in[26];
    const float* hrw3 = (const float*)d_in[27], *hrb3 = (const float*)d_in[28];
    const float* taw1 = (const float*)d_in[29], *tab1 = (const float*)d_in[30];
    const float* taw2 = (const float*)d_in[31], *tab2 = (const float*)d_in[32];

    float* out_tucker = (float*)d_out;                       // [B, N]
    float* out_poss   = (float*)d_out + (size_t)BB * NN;     // [B, N]

    // Workspace carve-up (~70 MB of fp32)
    float* w = (float*)d_ws;
    size_t off = 0;
    float* bufT  = w + off; off += (size_t)NN * EE;   // hidden buffer (reused for every 512-wide hidden)
    float* ts    = w + off; off += (size_t)NN * CC;
    float* ta    = w + off; off += (size_t)NN * CC;   // -> tam (in place)
    float* hs    = w + off; off += (size_t)BB * CC;   // -> ha  (in place)
    float* rs    = w + off; off += (size_t)BB * CC;
    float* Wm    = w + off; off += (size_t)BB * CC;   // -> BN'd (in place)
    float* h1    = w + off; off += (size_t)BB * 2 * CC;
    float* h2    = w + off; off += (size_t)BB * 2 * CC;
    float* hra   = w + off; off += (size_t)BB * CC;   // -> hrm (in place)
    float* inter = w + off; off += (size_t)BB * CC;

    const dim3 blk(128);
    auto grid = [](int M, int Nc) { return dim3((Nc + BN - 1) / BN, (M + BM - 1) / BM); };

    // ----- TuckER branch -----
    // hs = relu(head @ hsw1 + b) @ hsw2 + b
    gemm_wmma_f32<0,0,true ,true ><<<grid(BB, EE), blk, 0, stream>>>(head, nullptr, hsw1, hsb1, bufT, BB, EE, EE, EE, EE, EE);
    gemm_wmma_f32<0,0,false,true ><<<grid(BB, CC), blk, 0, stream>>>(bufT, nullptr, hsw2, hsb2, hs,   BB, CC, EE, EE, CC, CC);
    // rs = relu(rel @ rsw1 + b) @ rsw2 + b
    gemm_wmma_f32<0,0,true ,true ><<<grid(BB, EE), blk, 0, stream>>>(relv, nullptr, rsw1, rsb1, bufT, BB, EE, EE, EE, EE, EE);
    gemm_wmma_f32<0,0,false,true ><<<grid(BB, CC), blk, 0, stream>>>(bufT, nullptr, rsw2, rsb2, rs,   BB, CC, EE, EE, CC, CC);
    // ha = BN0(hs)  (in place)
    bn_kernel<<<CC, 256, 0, stream>>>(hs, hs, bn0g, bn0b, BB, CC);
    // ts = relu(tail @ tsw1 + b) @ tsw2 + b
    gemm_wmma_f32<0,0,true ,true ><<<grid(NN, EE), blk, 0, stream>>>(tail, nullptr, tsw1, tsb1, bufT, NN, EE, EE, EE, EE, EE);
    gemm_wmma_f32<0,0,false,true ><<<grid(NN, CC), blk, 0, stream>>>(bufT, nullptr, tsw2, tsb2, ts,   NN, CC, EE, EE, CC, CC);
    // Wm[b,d] = sum_{k,c} rs[b,k]*ha[b,c]*core[k,c,d]  == outer(rs,ha) @ core.reshape(C*C, C)
    gemm_wmma_f32<2,0,false,false><<<grid(BB, CC), blk, 0, stream>>>(rs, hs, core, nullptr, Wm, BB, CC, CC * CC, 0, CC, CC);
    // Wm = BN1(Wm)  (in place)
    bn_kernel<<<CC, 256, 0, stream>>>(Wm, Wm, bn1g, bn1b, BB, CC);
    // tucker_logits = Wm @ ts^T
    gemm_wmma_f32<0,1,false,false><<<grid(BB, NN), blk, 0, stream>>>(Wm, nullptr, ts, nullptr, out_tucker, BB, NN, CC, CC, CC, NN);

    // ----- possibility-codebook branch -----
    // hra = MLP3(concat(head, rel))
    gemm_wmma_f32<1,0,true ,true ><<<grid(BB, 2*CC), blk, 0, stream>>>(head, relv, hrw1, hrb1, h1, BB, 2*CC, 2*EE, 0,    2*CC, 2*CC);
    gemm_wmma_f32<0,0,true ,true ><<<grid(BB, 2*CC), blk, 0, stream>>>(h1, nullptr, hrw2, hrb2, h2, BB, 2*CC, 2*CC, 2*CC, 2*CC, 2*CC);
    gemm_wmma_f32<0,0,false,true ><<<grid(BB,   CC), blk, 0, stream>>>(h2, nullptr, hrw3, hrb3, hra, BB, CC, 2*CC, 2*CC, CC, CC);
    // ta = relu(tail @ taw1 + b) @ taw2 + b
    gemm_wmma_f32<0,0,true ,true ><<<grid(NN, EE), blk, 0, stream>>>(tail, nullptr, taw1, tab1, bufT, NN, EE, EE, EE, EE, EE);
    gemm_wmma_f32<0,0,false,true ><<<grid(NN, CC), blk, 0, stream>>>(bufT, nullptr, taw2, tab2, ta,   NN, CC, EE, EE, CC, CC);
    // hrm, tam = soft-topk masks (in place)
    topk_mask_kernel<<<BB, 128, 0, stream>>>(hra, hra);
    topk_mask_kernel<<<NN, 128, 0, stream>>>(ta, ta);
    // inter = einsum('bc,bcd->bd', hrm, tanh(codebook[rel]))
    inter_kernel<<<BB, 128, 0, stream>>>(hra, ridx, codebook, inter);
    // possibility_score = inter @ tam^T
    gemm_wmma_f32<0,1,false,false><<<grid(BB, NN), blk, 0, stream>>>(inter, nullptr, ta, nullptr, out_poss, BB, NN, CC, CC, CC, NN);
}